// VGAE_51823075393706
// MI455X (gfx1250) — compile-verified
//
#include <hip/hip_runtime.h>

typedef float v2f __attribute__((ext_vector_type(2)));
typedef float v8f __attribute__((ext_vector_type(8)));

static constexpr int NN = 100000;   // nodes

// ---------------- degree / normalization ----------------

__global__ void k_fill1(float* __restrict__ p, int n) {
  int i = blockIdx.x * blockDim.x + threadIdx.x;
  if (i < n) p[i] = 1.0f;                       // self-loop contributes 1 to deg
}

__global__ void k_count(const int* __restrict__ col, float* __restrict__ deg, int E) {
  int i = blockIdx.x * blockDim.x + threadIdx.x;
  if (i < E) atomicAdd(&deg[col[i]], 1.0f);
}

__global__ void k_rsqrt_ip(float* __restrict__ d, int n) {
  int i = blockIdx.x * blockDim.x + threadIdx.x;
  if (i < n) d[i] = rsqrtf(d[i]);               // deg >= 1 always (self loops)
}

// ---------------- fp32 WMMA GEMM: C[M x N] = A[M x K] * B[K x N] ----------------
// One wave computes a 16x16 output tile via V_WMMA_F32_16X16X4_F32, K fully unrolled.
// Layouts per CDNA5 ISA 7.12.2:
//   A (16x4):  lane L -> m = L%16, VGPR v -> k = 2*(L/16)+v   (contiguous float2)
//   B (4x16):  lane L -> n = L%16, VGPR v -> k = 2*(L/16)+v
//   C (16x16): lane L -> n = L%16, VGPR v -> m = v + 8*(L/16)
template <int K, int N>
__global__ __launch_bounds__(256) void k_gemm_wmma(const float* __restrict__ A,
                                                   const float* __restrict__ B,
                                                   float* __restrict__ C, int mtiles) {
  const int wave = threadIdx.x >> 5;
  const int lane = threadIdx.x & 31;
  const int mt = blockIdx.x * (blockDim.x >> 5) + wave;
  if (mt >= mtiles) return;                     // wave-uniform exit; EXEC all-1 otherwise
  const int row0 = mt << 4;
  const int col0 = blockIdx.y << 4;
  const int mn = lane & 15;                     // A row / B col within tile
  const int g  = lane >> 4;                     // lane group (k offset 0 or 2)

  v8f acc = {};
  const float* ap = A + (size_t)(row0 + mn) * K + 2 * g;
  const float* bp = B + (size_t)(2 * g) * N + col0 + mn;

#pragma unroll
  for (int k0 = 0; k0 < K; k0 += 4) {
    v2f a = *(const v2f*)(ap + k0);             // 8B-aligned: k0+2g is even
    v2f b;
    b.x = bp[(size_t)k0 * N];
    b.y = bp[(size_t)(k0 + 1) * N];
    acc = __builtin_amdgcn_wmma_f32_16x16x4_f32(false, a, false, b, (short)0, acc,
                                                false, false);
  }

  float* cp = C + (size_t)row0 * N + col0 + mn;
#pragma unroll
  for (int v = 0; v < 8; ++v) cp[(size_t)(v + 8 * g) * N] = acc[v];
}

// ---------------- GCN aggregation ----------------
// out[i,f] = dis[i]^2 * P[i,f] + bias[f]   (self-loop term + bias, also initializes out)
template <int F>
__global__ void k_selfbias(float* __restrict__ out, const float* __restrict__ P,
                           const float* __restrict__ dis, const float* __restrict__ bias) {
  int i = blockIdx.x * blockDim.x + threadIdx.x;
  if (i < NN * F) {
    int node = i / F;
    int f = i - node * F;
    float d = dis[node];
    out[i] = d * d * P[i] + bias[f];
  }
}

// Edge scatter, F=128: 32 lanes per edge, float4 per lane, 4 float atomics.
__global__ void k_scatter_f128(float* __restrict__ out, const float* __restrict__ P,
                               const float* __restrict__ dis,
                               const int* __restrict__ row, const int* __restrict__ col,
                               int E) {
  int gid = blockIdx.x * blockDim.x + threadIdx.x;
  int e = gid >> 5;
  if (e >= E) return;
  int f = (gid & 31) << 2;
  int r = row[e], c = col[e];
  float w = dis[r] * dis[c];
  const float4 p = *(const float4*)(P + (size_t)r * 128 + f);
  float* o = out + (size_t)c * 128 + f;
  atomicAdd(o + 0, w * p.x);
  atomicAdd(o + 1, w * p.y);
  atomicAdd(o + 2, w * p.z);
  atomicAdd(o + 3, w * p.w);
}

// Fused edge scatter for mu & logvar (F=64 each): 16 lanes/edge, shares index + norm.
__global__ void k_scatter_f64x2(float* __restrict__ outA, float* __restrict__ outB,
                                const float* __restrict__ PA, const float* __restrict__ PB,
                                const float* __restrict__ dis,
                                const int* __restrict__ row, const int* __restrict__ col,
                                int E) {
  int gid = blockIdx.x * blockDim.x + threadIdx.x;
  int e = gid >> 4;
  if (e >= E) return;
  int f = (gid & 15) << 2;
  int r = row[e], c = col[e];
  float w = dis[r] * dis[c];
  const float4 pa = *(const float4*)(PA + (size_t)r * 64 + f);
  const float4 pb = *(const float4*)(PB + (size_t)r * 64 + f);
  float* oa = outA + (size_t)c * 64 + f;
  float* ob = outB + (size_t)c * 64 + f;
  atomicAdd(oa + 0, w * pa.x);
  atomicAdd(oa + 1, w * pa.y);
  atomicAdd(oa + 2, w * pa.z);
  atomicAdd(oa + 3, w * pa.w);
  atomicAdd(ob + 0, w * pb.x);
  atomicAdd(ob + 1, w * pb.y);
  atomicAdd(ob + 2, w * pb.z);
  atomicAdd(ob + 3, w * pb.w);
}

__global__ void k_relu(float* __restrict__ h, int n) {
  int i = blockIdx.x * blockDim.x + threadIdx.x;
  if (i < n) h[i] = fmaxf(h[i], 0.0f);
}

__global__ void k_z(float* __restrict__ z, const float* __restrict__ mu,
                    const float* __restrict__ lv, const float* __restrict__ eps, int n) {
  int i = blockIdx.x * blockDim.x + threadIdx.x;
  if (i < n) z[i] = mu[i] + eps[i] * expf(0.5f * lv[i]);
}

// ---------------- launcher ----------------

extern "C" void kernel_launch(void* const* d_in, const int* in_sizes, int n_in,
                              void* d_out, int out_size, void* d_ws, size_t ws_size,
                              hipStream_t stream) {
  const float* x   = (const float*)d_in[0];
  const int*   ei  = (const int*)d_in[1];   // edge_index [2, E] (int32 per harness)
  const float* W1  = (const float*)d_in[2];
  const float* b1  = (const float*)d_in[3];
  const float* Wmu = (const float*)d_in[4];
  const float* bmu = (const float*)d_in[5];
  const float* Wlv = (const float*)d_in[6];
  const float* blv = (const float*)d_in[7];
  const float* eps = (const float*)d_in[8];

  const int E = in_sizes[1] / 2;
  const int* row = ei;
  const int* col = ei + E;

  // Workspace layout (floats): dis | P (reused as M1|L1) | H   ~= 103 MB
  float* ws  = (float*)d_ws;
  float* dis = ws;                              // NN
  float* P   = ws + 102400;                     // NN*128 (aligned start)
  float* H   = P + (size_t)NN * 128;            // NN*128
  float* M1  = P;                               // reuse after H is built
  float* L1  = P + (size_t)NN * 64;

  float* zout  = (float*)d_out;                 // [NN,64]
  float* muout = zout + (size_t)NN * 64;        // [NN,64]
  float* lvout = zout + (size_t)NN * 128;       // [NN,64]

  const int B = 256;
  const int mtiles = NN / 16;                   // 6250, exact
  dim3 g128((mtiles + 7) / 8, 128 / 16);
  dim3 g64((mtiles + 7) / 8, 64 / 16);

  // 1) symmetric normalization dis = deg^{-1/2}
  k_fill1<<<(NN + B - 1) / B, B, 0, stream>>>(dis, NN);
  k_count<<<(E + B - 1) / B, B, 0, stream>>>(col, dis, E);
  k_rsqrt_ip<<<(NN + B - 1) / B, B, 0, stream>>>(dis, NN);

  // 2) layer 1: P = x @ W1 ; H = relu(aggregate(P) + b1)
  k_gemm_wmma<128, 128><<<g128, B, 0, stream>>>(x, W1, P, mtiles);
  k_selfbias<128><<<(NN * 128 + B - 1) / B, B, 0, stream>>>(H, P, dis, b1);
  k_scatter_f128<<<(unsigned)(((size_t)E * 32 + B - 1) / B), B, 0, stream>>>(H, P, dis, row, col, E);
  k_relu<<<(NN * 128 + B - 1) / B, B, 0, stream>>>(H, NN * 128);

  // 3) mu / logvar heads
  k_gemm_wmma<128, 64><<<g64, B, 0, stream>>>(H, Wmu, M1, mtiles);
  k_gemm_wmma<128, 64><<<g64, B, 0, stream>>>(H, Wlv, L1, mtiles);
  k_selfbias<64><<<(NN * 64 + B - 1) / B, B, 0, stream>>>(muout, M1, dis, bmu);
  k_selfbias<64><<<(NN * 64 + B - 1) / B, B, 0, stream>>>(lvout, L1, dis, blv);
  k_scatter_f64x2<<<(unsigned)(((size_t)E * 16 + B - 1) / B), B, 0, stream>>>(
      muout, lvout, M1, L1, dis, row, col, E);

  // 4) reparameterize
  k_z<<<(NN * 64 + B - 1) / B, B, 0, stream>>>(zout, muout, lvout, eps, NN * 64);
}